// QuantumEntangler_13065290514840
// MI455X (gfx1250) — compile-verified
//
#include <hip/hip_runtime.h>

#define NQ 6
#define DEPTH 2

typedef float v2f __attribute__((ext_vector_type(2)));
typedef float v8f __attribute__((ext_vector_type(8)));

struct C2 { float re, im; };
__device__ __forceinline__ C2 cmul(C2 a, C2 b) { return {a.re*b.re - a.im*b.im, a.re*b.im + a.im*b.re}; }
__device__ __forceinline__ C2 cadd(C2 a, C2 b) { return {a.re + b.re, a.im + b.im}; }

// ---------------------------------------------------------------------------
// Kernel 1: build the 64x64 complex entangler unitary (batch-independent).
// Thread c simulates the circuit on basis state |c>, i.e. computes column c.
// Result written to workspace in WMMA-B "K-paired" layout:
//   ws[(k>>1)*128 + 2*n + (k&1)] = U[n][k]   (re), +4096 floats for im
// so kernel 2 loads each B fragment as one aligned float2 per lane.
// ---------------------------------------------------------------------------
__global__ __launch_bounds__(64) void build_unitary_kernel(
    const float* __restrict__ theta, const float* __restrict__ phi,
    const float* __restrict__ omega, float* __restrict__ wsU)
{
  __shared__ __align__(16) float sre[64][64];
  __shared__ __align__(16) float sim[64][64];
  const int c = threadIdx.x;

  for (int j = 0; j < 64; ++j) { sre[c][j] = (j == c) ? 1.0f : 0.0f; sim[c][j] = 0.0f; }

  for (int l = 0; l < DEPTH; ++l) {
    // Per-wire rotations: state = RX(om) * RY(ph) * RZ(th) * state
    for (int w = 0; w < NQ; ++w) {
      const int p = 5 - w;  // wire 0 = MSB
      const float th = theta[l*NQ + w] * 0.5f;
      const float ph = phi[l*NQ + w] * 0.5f;
      const float om = omega[l*NQ + w] * 0.5f;
      float st, ct, sy, cy, sx, cx;
      __sincosf(th, &st, &ct);
      __sincosf(ph, &sy, &cy);
      __sincosf(om, &sx, &cx);
      const C2 em = {ct, -st};   // e^{-i th/2}
      const C2 ep = {ct,  st};   // e^{+i th/2}
      // G = RX * RY * RZ (2x2 complex), fused
      const C2 g00 = cmul(em, { cx*cy, -sx*sy});
      const C2 g01 = cmul(ep, {-cx*sy, -sx*cy});
      const C2 g10 = cmul(em, { cx*sy, -sx*cy});
      const C2 g11 = cmul(ep, { cx*cy,  sx*sy});
      for (int t = 0; t < 32; ++t) {
        const int lo = t & ((1 << p) - 1);
        const int j0 = ((t >> p) << (p + 1)) | lo;
        const int j1 = j0 | (1 << p);
        const C2 a0 = {sre[c][j0], sim[c][j0]};
        const C2 a1 = {sre[c][j1], sim[c][j1]};
        const C2 n0 = cadd(cmul(g00, a0), cmul(g01, a1));
        const C2 n1 = cadd(cmul(g10, a0), cmul(g11, a1));
        sre[c][j0] = n0.re; sim[c][j0] = n0.im;
        sre[c][j1] = n1.re; sim[c][j1] = n1.im;
      }
    }
    // CNOT ring: control w, target (w+1)%NQ
    for (int w = 0; w < NQ; ++w) {
      const int pc = 5 - w;
      const int pt = 5 - ((w + 1) % NQ);
      for (int j = 0; j < 64; ++j) {
        if (((j >> pc) & 1) && !((j >> pt) & 1)) {
          const int j2 = j | (1 << pt);
          float t0 = sre[c][j]; sre[c][j] = sre[c][j2]; sre[c][j2] = t0;
          float t1 = sim[c][j]; sim[c][j] = sim[c][j2]; sim[c][j2] = t1;
        }
      }
    }
  }

  // Write out in K-paired B layout (column c is K index c; row j is N index j).
  const int base = (c >> 1) * 128 + (c & 1);
  for (int j = 0; j < 64; ++j) {
    wsU[base + 2*j]        = sre[c][j];
    wsU[4096 + base + 2*j] = sim[c][j];
  }
}

// ---------------------------------------------------------------------------
// Kernel 2: per 16-row batch tile (one wave each), compute encoded state,
// two f32 WMMA GEMM chains (re/im), probabilities, <Z_w>, gathered output.
// Block = 128 threads = 4 waves = 64 batch rows.
// ---------------------------------------------------------------------------
__global__ __launch_bounds__(128) void qcirc_main_kernel(
    const float* __restrict__ x, const float* __restrict__ wsU,
    float* __restrict__ out, int Bsz)
{
  __shared__ __align__(16) float uB[8192];        // re[4096] | im[4096], K-paired layout
  __shared__ __align__(16) float s0[4][16][64];   // per-wave A tile; reused for p

  const int tid  = threadIdx.x;
  const int wave = tid >> 5;
  const int lane = tid & 31;
  const int h    = lane >> 4;    // K/N half selector
  const int m    = lane & 15;    // row-in-tile / col-in-tile
  const int r0   = blockIdx.x * 64 + wave * 16;

  // Stage the unitary into LDS (whole workgroup).
  for (int i = tid; i < 8192; i += 128) uB[i] = wsU[i];

  // Encoded product state: lane pair (m, m+16) covers row m, j-halves [0,32) and [32,64).
  int row = r0 + m;
  if (row >= Bsz) row = Bsz - 1;  // clamp; stores are guarded below
  float cw[NQ], sw[NQ];
  #pragma unroll
  for (int w = 0; w < NQ; ++w) {
    const float f = tanhf(x[row * NQ + w]) * 3.14159265358979323846f;
    __sincosf(0.5f * f, &sw[w], &cw[w]);
  }
  const int jb = h * 32;
  for (int j32 = 0; j32 < 32; ++j32) {
    const int j = jb + j32;
    float prod = 1.0f;
    #pragma unroll
    for (int w = 0; w < NQ; ++w) prod *= ((j >> (5 - w)) & 1) ? sw[w] : cw[w];
    s0[wave][m][j] = prod;
  }
  __syncthreads();

  // Hoist all 16 A fragments (16x4 f32 each) into registers.
  // A layout: lane holds M=m, K = k0 + 2*h + {0,1}  for k0 = 4*ks.
  v2f afrag[16];
  #pragma unroll
  for (int ks = 0; ks < 16; ++ks) {
    const int k = ks * 4 + 2 * h;
    afrag[ks] = *(const v2f*)&s0[wave][m][k];
  }

  const float* uRe = uB;
  const float* uIm = uB + 4096;

  // GEMM + |.|^2 epilogue, tile-by-tile over N.
  for (int n0t = 0; n0t < 4; ++n0t) {
    const int nn = n0t * 16 + m;  // B-fragment N index for this lane
    v8f ar = {0.f,0.f,0.f,0.f,0.f,0.f,0.f,0.f};
    v8f ai = {0.f,0.f,0.f,0.f,0.f,0.f,0.f,0.f};
    #pragma unroll
    for (int ks = 0; ks < 16; ++ks) {
      const int bidx = (ks * 2 + h) * 128 + 2 * nn;   // K = 4*ks + 2*h + {0,1}
      const v2f br = *(const v2f*)&uRe[bidx];
      const v2f bi = *(const v2f*)&uIm[bidx];
      ar = __builtin_amdgcn_wmma_f32_16x16x4_f32(false, afrag[ks], false, br,
                                                 (short)0, ar, false, false);
      ai = __builtin_amdgcn_wmma_f32_16x16x4_f32(false, afrag[ks], false, bi,
                                                 (short)0, ai, false, false);
    }
    // p = re^2 + im^2 ; D layout: row = r + 8*h, col = nn. Reuse s0 buffer.
    #pragma unroll
    for (int r = 0; r < 8; ++r) {
      const float pv = ar[r] * ar[r] + ai[r] * ai[r];
      s0[wave][r + 8 * h][nn] = pv;
    }
  }
  __syncthreads();

  // z[w] = sum_j p[m][j] * (1 - 2*bit_w(j)); lane pair splits j, combine via shfl.
  float z[NQ] = {0.f, 0.f, 0.f, 0.f, 0.f, 0.f};
  for (int j32 = 0; j32 < 32; ++j32) {
    const int j = jb + j32;
    const float pv = s0[wave][m][j];
    #pragma unroll
    for (int w = 0; w < NQ; ++w) z[w] += ((j >> (5 - w)) & 1) ? -pv : pv;
  }
  #pragma unroll
  for (int w = 0; w < NQ; ++w) z[w] += __shfl_xor(z[w], 16, 32);

  if (lane < 16) {
    const int orow = r0 + m;
    if (orow < Bsz) {
      float* o = out + (size_t)orow * 8;
      #pragma unroll
      for (int k = 0; k < 8; ++k) o[k] = z[k % NQ];
    }
  }
}

extern "C" void kernel_launch(void* const* d_in, const int* in_sizes, int n_in,
                              void* d_out, int out_size, void* d_ws, size_t ws_size,
                              hipStream_t stream) {
  const float* x     = (const float*)d_in[0];
  const float* theta = (const float*)d_in[1];
  const float* phi   = (const float*)d_in[2];
  const float* omega = (const float*)d_in[3];
  float* out = (float*)d_out;
  float* ws  = (float*)d_ws;   // needs 8192 floats = 32 KB
  const int Bsz = in_sizes[0] / NQ;

  build_unitary_kernel<<<1, 64, 0, stream>>>(theta, phi, omega, ws);
  const int blocks = (Bsz + 63) / 64;
  qcirc_main_kernel<<<blocks, 128, 0, stream>>>(x, ws, out, Bsz);
}